// SinkhornSpectralLoss_86852828660259
// MI455X (gfx1250) — compile-verified
//
#include <hip/hip_runtime.h>
#include <hip/hip_bf16.h>

#define NB 513          // NBINS
#define NROWS 504       // 8 clips * 63 frames
#define MPAD 512
#define FFTN 1024
#define HOP 512
#define NTILE 33        // ceil(513/16)
#define MTILE 32        // 512/16
#define EPS_S 0.01f
#define IEPS_S 100.0f
#define PI_F 3.14159265358979323846f

typedef __attribute__((ext_vector_type(2))) float v2f;
typedef __attribute__((ext_vector_type(8))) float v8f;

// ---------------------------------------------------------------- reductions
__device__ __forceinline__ float block_sum(float val, float* sred) {
    // wave32 shuffle reduce, then cross-wave via LDS
    #pragma unroll
    for (int off = 16; off > 0; off >>= 1) val += __shfl_down(val, off, 32);
    int lane = threadIdx.x & 31, wid = threadIdx.x >> 5;
    if (lane == 0) sred[wid] = val;
    __syncthreads();
    if (threadIdx.x == 0) {
        float t = 0.f;
        int nw = (blockDim.x + 31) >> 5;
        for (int i = 0; i < nw; ++i) t += sred[i];
        sred[0] = t;
    }
    __syncthreads();
    float r = sred[0];
    __syncthreads();
    return r;
}

// ------------------------------------------------- 1) windowed frame matrix
// frames[input][m][n], m in [0,512), rows >= 504 zeroed (GEMM padding)
__global__ void frames_kernel(const float* __restrict__ xa,
                              const float* __restrict__ ta,
                              float* __restrict__ fx, float* __restrict__ ft) {
    int idx = blockIdx.x * blockDim.x + threadIdx.x;
    if (idx >= 2 * MPAD * FFTN) return;
    int which = idx >> 19;            // 512*1024 = 2^19
    int rem = idx & ((MPAD * FFTN) - 1);
    int m = rem >> 10;
    int n = rem & (FFTN - 1);
    const float* a = which ? ta : xa;
    float* f = which ? ft : fx;
    float val = 0.f;
    if (m < NROWS) {
        int b = m / 63, fr = m % 63;
        // periodic hann: 0.5*(1 - cos(2*pi*n/1024)) = 0.5*(1 - cos(n*pi/512))
        float w = 0.5f * (1.0f - __cosf((float)n * (PI_F / 512.0f)));
        val = a[b * 32768 + fr * HOP + n] * w;
    }
    f[rem] = val;
}

// --------------------------------------- 2) DFT-as-GEMM via f32 WMMA 16x16x4
// One wave computes one 16x16 output tile holding BOTH Re (cos) and Im (sin)
// accumulators. Basis generated on the fly with exact mod-1024 phase.
__global__ __launch_bounds__(256) void spectro_kernel(const float* __restrict__ fx,
                                                      const float* __restrict__ ft,
                                                      float* __restrict__ sx,
                                                      float* __restrict__ st) {
    int lane = threadIdx.x & 31;
    int widx = threadIdx.x >> 5;
    int w = blockIdx.x * 8 + widx;              // 264 blocks * 8 waves = 2112 tiles
    int input = w / (MTILE * NTILE);
    int tile = w % (MTILE * NTILE);
    int tm = tile & (MTILE - 1);
    int tn = tile >> 5;                          // 0..32
    const float* F = input ? ft : fx;
    float* S = input ? st : sx;

    int lr = lane & 15;
    int khalf = (lane < 16) ? 0 : 2;             // A/B frag: lanes 0-15 hold K={0,1}, 16-31 hold K={2,3}
    int m = tm * 16 + lr;                        // A row for this lane
    int j = tn * 16 + lr;                        // B column (frequency bin)
    const float* arow = F + m * FFTN + khalf;

    v8f accr = {0.f, 0.f, 0.f, 0.f, 0.f, 0.f, 0.f, 0.f};
    v8f acci = {0.f, 0.f, 0.f, 0.f, 0.f, 0.f, 0.f, 0.f};

    for (int kk = 0; kk < FFTN / 4; ++kk) {
        int kbase = kk * 4 + khalf;
        v2f a = *(const v2f*)(arow + kk * 4);
        // exact phase: (j*k) mod 1024, angle = p * 2pi/1024
        int p0 = (j * kbase) & (FFTN - 1);
        int p1 = (p0 + j) & (FFTN - 1);
        float c0, s0, c1, s1;
        __sincosf((float)p0 * (PI_F / 512.0f), &s0, &c0);
        __sincosf((float)p1 * (PI_F / 512.0f), &s1, &c1);
        v2f bc = {c0, c1};
        v2f bs = {s0, s1};
        accr = __builtin_amdgcn_wmma_f32_16x16x4_f32(false, a, false, bc,
                                                     (short)0, accr, false, false);
        acci = __builtin_amdgcn_wmma_f32_16x16x4_f32(false, a, false, bs,
                                                     (short)0, acci, false, false);
    }

    // D layout: VGPR r -> M = r (lanes 0-15) or 8+r (lanes 16-31); N = lane%16
    int mbase = tm * 16 + ((lane < 16) ? 0 : 8);
    if (j < NB) {
        #pragma unroll
        for (int r = 0; r < 8; ++r) {
            int mm = mbase + r;
            if (mm < NROWS) {
                float re = accr[r], im = acci[r];
                S[mm * NB + j] = __logf(re * re + im * im + 1e-5f);
            }
        }
    }
}

// ---------------------------------------- 3) per-row normalize -> log-probs
__global__ __launch_bounds__(544) void normalize_kernel(const float* __restrict__ spec,
                                                        float* __restrict__ mu) {
    __shared__ float sred[32];
    int row = blockIdx.x;
    int t = threadIdx.x;
    float xc = 0.f;
    if (t < NB) {
        float x = spec[row * NB + t];
        x = fminf(fmaxf(x, 0.f), 1e9f);   // clamp (value path of stop-gradient trick)
        xc = x + 1e-9f;
    }
    float s = block_sum(xc, sred);
    if (t < NB) mu[row * NB + t] = __logf(xc) - __logf(s);
}

// ------------------------------------------------- 4) Sinkhorn, block per row
__global__ __launch_bounds__(544) void sinkhorn_kernel(const float* __restrict__ mu_g,
                                                       const float* __restrict__ nu_g,
                                                       float* __restrict__ acc) {
    __shared__ float mu[NB], nu[NB], u[NB], v[NB];
    __shared__ float sred[32];
    int row = blockIdx.x;
    int t = threadIdx.x;
    if (t < NB) {
        mu[t] = mu_g[row * NB + t];
        nu[t] = nu_g[row * NB + t];
        u[t] = 0.f;
        v[t] = 0.f;
    }
    __syncthreads();
    const float cstep = 1.0f / 512.0f;

    for (int it = 0; it < 5; ++it) {
        // u_i = eps*mu_i - eps*LSE_j[(v_j - C_ij)/eps]   (old u cancels exactly)
        float unew = 0.f;
        if (t < NB) {
            float mx = -1e30f;
            for (int jj = 0; jj < NB; ++jj) {
                float x = v[jj] - fabsf((float)(t - jj)) * cstep;
                mx = fmaxf(mx, x);
            }
            float s = 0.f;
            for (int jj = 0; jj < NB; ++jj) {
                float x = v[jj] - fabsf((float)(t - jj)) * cstep;
                s += __expf((x - mx) * IEPS_S);
            }
            unew = EPS_S * mu[t] - mx - EPS_S * __logf(s);
        }
        __syncthreads();
        if (t < NB) u[t] = unew;
        __syncthreads();
        // v_j = eps*nu_j - eps*LSE_i[(u_i - C_ij)/eps]  (C symmetric)
        float vnew = 0.f;
        if (t < NB) {
            float mx = -1e30f;
            for (int ii = 0; ii < NB; ++ii) {
                float x = u[ii] - fabsf((float)(t - ii)) * cstep;
                mx = fmaxf(mx, x);
            }
            float s = 0.f;
            for (int ii = 0; ii < NB; ++ii) {
                float x = u[ii] - fabsf((float)(t - ii)) * cstep;
                s += __expf((x - mx) * IEPS_S);
            }
            vnew = EPS_S * nu[t] - mx - EPS_S * __logf(s);
        }
        __syncthreads();
        if (t < NB) v[t] = vnew;
        __syncthreads();
    }

    // cost = sum_ij exp((u_i + v_j - C_ij)/eps) * C_ij
    float c = 0.f;
    if (t < NB) {
        float ui = u[t];
        for (int jj = 0; jj < NB; ++jj) {
            float C = fabsf((float)(t - jj)) * cstep;
            c += __expf((ui + v[jj] - C) * IEPS_S) * C;
        }
    }
    float tot = block_sum(c, sred);
    if (t == 0) atomicAdd(acc, tot);
}

// ------------------------------------------------------------- init/finalize
__global__ void zero_kernel(float* acc) {
    if (threadIdx.x == 0 && blockIdx.x == 0) *acc = 0.f;
}
__global__ void final_kernel(const float* __restrict__ acc, float* __restrict__ out) {
    if (threadIdx.x == 0 && blockIdx.x == 0) out[0] = acc[0] * (1.0f / (float)NROWS);
}

// ---------------------------------------------------------------------------
extern "C" void kernel_launch(void* const* d_in, const int* in_sizes, int n_in,
                              void* d_out, int out_size, void* d_ws, size_t ws_size,
                              hipStream_t stream) {
    const float* xa = (const float*)d_in[0];
    const float* ta = (const float*)d_in[1];
    float* ws = (float*)d_ws;

    float* fx = ws;                       // 512*1024
    float* ft = fx + MPAD * FFTN;         // 512*1024
    float* sx = ft + MPAD * FFTN;         // 504*513 log-power spec (x)
    float* st = sx + NROWS * NB;          // 504*513 (target)
    float* mx = st + NROWS * NB;          // mu
    float* mt = mx + NROWS * NB;          // nu
    float* acc = mt + NROWS * NB;         // 1 float accumulator

    int nfr = 2 * MPAD * FFTN;
    frames_kernel<<<(nfr + 255) / 256, 256, 0, stream>>>(xa, ta, fx, ft);
    // 2 inputs * 32 m-tiles * 33 n-tiles = 2112 waves / 8 waves per block
    spectro_kernel<<<264, 256, 0, stream>>>(fx, ft, sx, st);
    normalize_kernel<<<NROWS, 544, 0, stream>>>(sx, mx);
    normalize_kernel<<<NROWS, 544, 0, stream>>>(st, mt);
    zero_kernel<<<1, 32, 0, stream>>>(acc);
    sinkhorn_kernel<<<NROWS, 544, 0, stream>>>(mx, mt, acc);
    final_kernel<<<1, 32, 0, stream>>>(acc, (float*)d_out);
}